// OptimizedDoubleEKVBlock_27144193311004
// MI455X (gfx1250) — compile-verified
//
#include <hip/hip_runtime.h>
#include <hip/hip_bf16.h>
#include <math.h>

// ---------------------------------------------------------------------------
// Constants from the reference
// ---------------------------------------------------------------------------
#define INV_PHI   13.333333333333334f   // 1 / (2*1.5*0.025)
#define VD_SHIFT  6.666666666666667f    // 0.5 * INV_PHI
#define G1        0.023570226039551585f // ALPHA*BASE_TIA/sqrt(288)
#define G2        0.016666666666666666f // ALPHA*BASE_TIA/sqrt(576)
#define BN_EPS    1e-5f

// Numerically stable softplus. Argument of the log is 1 + e^{-|z|} in (1,2]:
// no cancellation, so the fast native v_exp_f32/v_log_f32 pair is accurate
// enough here (error contribution to s^2 is ~1e-6 absolute, values are ~1e4).
__device__ __forceinline__ float softplusf(float z) {
    float a = fabsf(z);
    return fmaxf(z, 0.0f) + __logf(1.0f + __expf(-a));
}

// ---------------------------------------------------------------------------
// CDNA5 async global->LDS copy (ASYNCcnt-tracked, no VGPR round trip).
// lds_byte_off is the byte offset inside this workgroup's LDS allocation;
// dynamic (extern __shared__) LDS starts at offset 0 in these kernels.
// ---------------------------------------------------------------------------
__device__ __forceinline__ void ld_async_b128(unsigned lds_byte_off, const void* gptr) {
    asm volatile("global_load_async_to_lds_b128 %0, %1, off"
                 :: "v"(lds_byte_off), "v"(gptr)
                 : "memory");
}
__device__ __forceinline__ void wait_async0() {
    asm volatile("s_wait_asynccnt 0" ::: "memory");
}

// ---------------------------------------------------------------------------
// K0: mapper(x) -> m1    (131072 elems)
// ---------------------------------------------------------------------------
__global__ void k_mapper(const float* __restrict__ x,
                         const float* __restrict__ scale_p,
                         const float* __restrict__ bias_p,
                         float* __restrict__ m1, int n) {
    int i = blockIdx.x * blockDim.x + threadIdx.x;
    if (i >= n) return;
    float s = *scale_p, b = *bias_p;
    m1[i] = fminf(fmaxf(fmaf(x[i], s, b), 0.0f), 9.0f);
}

// ---------------------------------------------------------------------------
// K1: EKV conv1: C=32,k=3,stride=2,pad=1, in 32x32 -> out 16x16
// grid (64 o, 4 n), 256 threads = one output pixel each.
// Stages full image (32768 f32 = 128KB) into LDS via async b128 copies
// (2 workgroups / 320KB WGP), theta row via plain loads (overlaps async).
// ---------------------------------------------------------------------------
__global__ void k_ekv_conv1(const float* __restrict__ m1,
                            const float* __restrict__ theta,
                            float* __restrict__ out1) {
    extern __shared__ float smem[];
    float* sIn = smem;               // 32*32*32 floats, LDS bytes [0, 131072)
    float* sTh = smem + 32 * 1024;   // 288 floats
    const int o = blockIdx.x, n = blockIdx.y, tid = threadIdx.x;

    const float* xin = m1 + n * 32768;
    const float* th  = theta + o * 288;
    __builtin_prefetch(th, 0, 3);

    // async-stage 8192 x b128 (16B) chunks: 32 per thread
    for (int i = tid; i < 8192; i += 256)
        ld_async_b128((unsigned)(i * 16), (const void*)(xin + i * 4));
    // theta through normal path, overlapped with the async copies
    for (int i = tid; i < 288; i += 256) sTh[i] = th[i];
    wait_async0();
    __syncthreads();

    const int ho = tid >> 4, wo = tid & 15;
    float acc = 0.0f;
    int t = 0;
    for (int c = 0; c < 32; ++c) {
        const float* sch = sIn + c * 1024;
        #pragma unroll
        for (int kh = 0; kh < 3; ++kh) {
            const int ih = 2 * ho - 1 + kh;
            const bool okh = (ih >= 0) & (ih < 32);
            #pragma unroll
            for (int kw = 0; kw < 3; ++kw) {
                const int iw = 2 * wo - 1 + kw;
                float p = (okh & (iw >= 0) & (iw < 32)) ? sch[ih * 32 + iw] : 0.0f;
                float z1 = (p - sTh[t++]) * INV_PHI;
                float z2 = z1 - VD_SHIFT;
                float s1 = softplusf(z1);
                float s2 = softplusf(z2);
                acc = fmaf(s1, s1, acc);
                acc = fmaf(-s2, s2, acc);
            }
        }
    }
    out1[(n * 64 + o) * 256 + tid] = acc * G1;
}

// ---------------------------------------------------------------------------
// K2: EKV conv2: C=64,k=3,stride=1,pad=1, in 16x16 -> out 16x16
// Stages 64KB activation via async b128 + theta row in LDS.
// ---------------------------------------------------------------------------
__global__ void k_ekv_conv2(const float* __restrict__ m2,
                            const float* __restrict__ theta,
                            float* __restrict__ out2) {
    extern __shared__ float smem[];
    float* sIn = smem;               // 64*256 floats, LDS bytes [0, 65536)
    float* sTh = smem + 64 * 256;    // 576 floats
    const int o = blockIdx.x, n = blockIdx.y, tid = threadIdx.x;

    const float* xin = m2 + n * 16384;
    const float* th  = theta + o * 576;
    __builtin_prefetch(th, 0, 3);

    // async-stage 4096 x b128 chunks: 16 per thread
    for (int i = tid; i < 4096; i += 256)
        ld_async_b128((unsigned)(i * 16), (const void*)(xin + i * 4));
    for (int i = tid; i < 576; i += 256) sTh[i] = th[i];
    wait_async0();
    __syncthreads();

    const int ho = tid >> 4, wo = tid & 15;
    float acc = 0.0f;
    int t = 0;
    for (int c = 0; c < 64; ++c) {
        const float* sch = sIn + c * 256;
        #pragma unroll
        for (int kh = 0; kh < 3; ++kh) {
            const int ih = ho - 1 + kh;
            const bool okh = (ih >= 0) & (ih < 16);
            #pragma unroll
            for (int kw = 0; kw < 3; ++kw) {
                const int iw = wo - 1 + kw;
                float p = (okh & (iw >= 0) & (iw < 16)) ? sch[ih * 16 + iw] : 0.0f;
                float z1 = (p - sTh[t++]) * INV_PHI;
                float z2 = z1 - VD_SHIFT;
                float s1 = softplusf(z1);
                float s2 = softplusf(z2);
                acc = fmaf(s1, s1, acc);
                acc = fmaf(-s2, s2, acc);
            }
        }
    }
    out2[(n * 64 + o) * 256 + tid] = acc * G2;
}

// ---------------------------------------------------------------------------
// K3: per-channel sum / sumsq over (N,H,W): buf is (4,64,16,16)
// ---------------------------------------------------------------------------
__global__ void k_chan_stats(const float* __restrict__ buf,
                             float* __restrict__ stats) {
    __shared__ float s1[256];
    __shared__ float s2[256];
    const int c = blockIdx.x, t = threadIdx.x;
    float a = 0.0f, b = 0.0f;
    #pragma unroll
    for (int n = 0; n < 4; ++n) {
        float v = buf[n * 16384 + c * 256 + t];
        a += v;
        b = fmaf(v, v, b);
    }
    s1[t] = a; s2[t] = b;
    __syncthreads();
    for (int s = 128; s > 0; s >>= 1) {
        if (t < s) { s1[t] += s1[t + s]; s2[t] += s2[t + s]; }
        __syncthreads();
    }
    if (t == 0) { stats[c * 2] = s1[0]; stats[c * 2 + 1] = s2[0]; }
}

// ---------------------------------------------------------------------------
// K4: m2 = mapper(BN(out1))
// ---------------------------------------------------------------------------
__global__ void k_bn_mapper(const float* __restrict__ in,
                            const float* __restrict__ stats,
                            const float* __restrict__ gamma,
                            const float* __restrict__ beta,
                            const float* __restrict__ scale_p,
                            const float* __restrict__ bias_p,
                            float* __restrict__ out) {
    int i = blockIdx.x * blockDim.x + threadIdx.x;   // 65536
    int c = (i >> 8) & 63;
    float mean = stats[2 * c] * (1.0f / 1024.0f);
    float var  = stats[2 * c + 1] * (1.0f / 1024.0f) - mean * mean;
    float rs   = rsqrtf(var + BN_EPS);
    float xn   = fmaf((in[i] - mean) * rs, gamma[c], beta[c]);
    out[i] = fminf(fmaxf(fmaf(xn, *scale_p, *bias_p), 0.0f), 9.0f);
}

// ---------------------------------------------------------------------------
// K5: shortcut 1x1 stride-2 conv as an f32 WMMA GEMM.
//   M = 1024 output positions, K = 32 in-channels, N = 64 out-channels.
//   One wave (32 threads) per 16x16 C tile, 8 chained V_WMMA_F32_16X16X4_F32.
//   A layout (ISA 7.12.2, 16x4 f32): lane = M row; lanes 0-15 hold {K0,K1},
//   lanes 16-31 hold {K2,K3}. B (4x16): lane%16 = N col, same K split.
//   C/D: VGPR r -> row r (lanes 0-15) / r+8 (lanes 16-31), col = lane%16.
// ---------------------------------------------------------------------------
typedef __attribute__((ext_vector_type(2))) float v2f;
typedef __attribute__((ext_vector_type(8))) float v8f;

__global__ void k_shortcut_wmma(const float* __restrict__ x,
                                const float* __restrict__ w,
                                float* __restrict__ sc) {
    const int tm   = blockIdx.x;         // 0..63 : 16-row M tile
    const int tn   = blockIdx.y;         // 0..3  : 16-col N tile
    const int lane = threadIdx.x;        // wave32
    const int r    = lane & 15;
    const int hi   = lane >> 4;          // 0 => {K0,K1}, 1 => {K2,K3}

    // A-row: output position p -> input pixel x[n, c, 2*ho, 2*wo]
    const int p  = tm * 16 + r;
    const int n  = p >> 8, rem = p & 255, ho = rem >> 4, wo = rem & 15;
    const float* xb = x + (size_t)n * 32 * 1024 + (2 * ho) * 32 + (2 * wo);
    const int ocol = tn * 16 + r;        // B-column: out channel

    v8f acc = {};
    #pragma unroll
    for (int k0 = 0; k0 < 32; k0 += 4) {
        const int ka = k0 + hi * 2;
        v2f a, b;
        a.x = xb[(size_t)ka * 1024];
        a.y = xb[(size_t)(ka + 1) * 1024];
        b.x = w[ocol * 32 + ka];
        b.y = w[ocol * 32 + ka + 1];
        acc = __builtin_amdgcn_wmma_f32_16x16x4_f32(
                  false, a, false, b, (short)0, acc, false, false);
    }

    #pragma unroll
    for (int i = 0; i < 8; ++i) {
        const int row = tm * 16 + i + hi * 8;          // global position
        const int n2 = row >> 8, rem2 = row & 255;
        const int ho2 = rem2 >> 4, wo2 = rem2 & 15;
        sc[((n2 * 64 + (tn * 16 + r)) * 16 + ho2) * 16 + wo2] = acc[i];
    }
}

// ---------------------------------------------------------------------------
// K6: out = relu(BN(out2) + BN(sc))
// ---------------------------------------------------------------------------
__global__ void k_final(const float* __restrict__ out2,
                        const float* __restrict__ st2,
                        const float* __restrict__ g2,
                        const float* __restrict__ b2,
                        const float* __restrict__ sc,
                        const float* __restrict__ stsc,
                        const float* __restrict__ gsc,
                        const float* __restrict__ bsc,
                        float* __restrict__ out) {
    int i = blockIdx.x * blockDim.x + threadIdx.x;   // 65536
    int c = (i >> 8) & 63;

    float m2 = st2[2 * c] * (1.0f / 1024.0f);
    float v2 = st2[2 * c + 1] * (1.0f / 1024.0f) - m2 * m2;
    float a  = fmaf((out2[i] - m2) * rsqrtf(v2 + BN_EPS), g2[c], b2[c]);

    float ms = stsc[2 * c] * (1.0f / 1024.0f);
    float vs = stsc[2 * c + 1] * (1.0f / 1024.0f) - ms * ms;
    float b  = fmaf((sc[i] - ms) * rsqrtf(vs + BN_EPS), gsc[c], bsc[c]);

    out[i] = fmaxf(a + b, 0.0f);
}

// ---------------------------------------------------------------------------
// Host-side orchestration
// ---------------------------------------------------------------------------
extern "C" void kernel_launch(void* const* d_in, const int* in_sizes, int n_in,
                              void* d_out, int out_size, void* d_ws, size_t ws_size,
                              hipStream_t stream) {
    (void)in_sizes; (void)n_in; (void)out_size; (void)ws_size;

    const float* x        = (const float*)d_in[0];
    const float* theta1   = (const float*)d_in[1];
    const float* theta2   = (const float*)d_in[2];
    const float* scale1   = (const float*)d_in[3];
    const float* bias1    = (const float*)d_in[4];
    const float* scale2   = (const float*)d_in[5];
    const float* bias2    = (const float*)d_in[6];
    const float* gamma1   = (const float*)d_in[7];
    const float* beta1    = (const float*)d_in[8];
    const float* gamma2   = (const float*)d_in[9];
    const float* beta2    = (const float*)d_in[10];
    const float* w_sc     = (const float*)d_in[11];
    const float* gamma_sc = (const float*)d_in[12];
    const float* beta_sc  = (const float*)d_in[13];
    float* out = (float*)d_out;

    // workspace layout (f32 elements)
    float* ws      = (float*)d_ws;
    float* m1      = ws;               // 131072
    float* out1    = ws + 131072;      // 65536
    float* m2      = ws + 196608;      // 65536
    float* out2    = ws + 262144;      // 65536
    float* sc      = ws + 327680;      // 65536
    float* stats1  = ws + 393216;      // 128
    float* stats2  = ws + 393344;      // 128
    float* statssc = ws + 393472;      // 128

    // 1) m1 = mapper(x)
    k_mapper<<<512, 256, 0, stream>>>(x, scale1, bias1, m1, 131072);

    // 2) out1 = ekv_conv(m1, theta1, stride=2, pad=1)
    k_ekv_conv1<<<dim3(64, 4), 256, (32 * 1024 + 288) * sizeof(float), stream>>>(
        m1, theta1, out1);

    // 3) BN stats of out1; m2 = mapper(BN(out1))
    k_chan_stats<<<64, 256, 0, stream>>>(out1, stats1);
    k_bn_mapper<<<256, 256, 0, stream>>>(out1, stats1, gamma1, beta1,
                                         scale2, bias2, m2);

    // 4) out2 = ekv_conv(m2, theta2, stride=1, pad=1); stats
    k_ekv_conv2<<<dim3(64, 4), 256, (64 * 256 + 576) * sizeof(float), stream>>>(
        m2, theta2, out2);
    k_chan_stats<<<64, 256, 0, stream>>>(out2, stats2);

    // 5) shortcut GEMM via f32 WMMA; stats
    k_shortcut_wmma<<<dim3(64, 4), 32, 0, stream>>>(x, w_sc, sc);
    k_chan_stats<<<64, 256, 0, stream>>>(sc, statssc);

    // 6) out = relu(BN(out2) + BN(sc))
    k_final<<<256, 256, 0, stream>>>(out2, stats2, gamma2, beta2,
                                     sc, statssc, gamma_sc, beta_sc, out);
}